// JustGraphStructureGeometric_16192026706672
// MI455X (gfx1250) — compile-verified
//
#include <hip/hip_runtime.h>
#include <hip/hip_bf16.h>

typedef float v2f __attribute__((ext_vector_type(2)));
typedef float v8f __attribute__((ext_vector_type(8)));

#define NN 50000
#define NE 600000
#define DIN 128
#define DH1 128
#define DH2 64
#define XPAD 4   // X strip row pad (floats): keeps 16B alignment + kills bank conflicts

static inline int cdiv(long long a, long long b) { return (int)((a + b - 1) / b); }

// ---------------- utility kernels ----------------

__global__ void k_fill(float* __restrict__ p, int n, float val) {
    int i = blockIdx.x * blockDim.x + threadIdx.x;
    if (i < n) p[i] = val;
}

__global__ void k_deg(const int* __restrict__ ei, float* __restrict__ deg) {
    int e = blockIdx.x * blockDim.x + threadIdx.x;
    if (e < NE) {
        int dst = ei[NE + e];
        unsafeAtomicAdd(&deg[dst], 1.0f);   // global_atomic_add_f32
    }
}

__global__ void k_dinv(float* __restrict__ deg) {
    int i = blockIdx.x * blockDim.x + threadIdx.x;
    if (i < NN) {
        float d = deg[i];
        deg[i] = (d > 0.0f) ? rsqrtf(d) : 0.0f;   // in place: deg -> deg^-1/2
    }
}

// ---------------- fp32 WMMA GEMM: C[M,BN] = A[M,KD] @ W[KD,BN] ----------------
// 256 threads = 8 waves; block covers 128 rows; wave w owns rows [w*16, w*16+16)
// and all BN columns via BN/16 v8f accumulators.
//  - W staged in LDS pre-swizzled into k-pairs so each B fragment is one ds_load_b64
//  - X strip staged via GLOBAL_LOAD_ASYNC_TO_LDS_B128 (ASYNCcnt path), rows padded
//  - B fragments loaded ascending, consumed DESCENDING: forces all fragments live
//    in distinct VGPRs so the backend batches the ds_loads and emits one dscnt wait
//    per 8-WMMA chain instead of one per 2.
template <int BN, int KD>
__global__ __launch_bounds__(256) void k_gemm_wmma(const float* __restrict__ A,
                                                   const float* __restrict__ W,
                                                   float* __restrict__ C, int M) {
    constexpr int WROW = 2 * BN + 2;   // dwords per k-pair row of swizzled W (8B aligned)
    constexpr int XROW = KD + XPAD;    // dwords per padded X row (16B aligned)
    extern __shared__ float smem[];
    float* sW = smem;                      // [(KD/2)][WROW]: sW[(k>>1)*WROW + 2n + (k&1)]
    float* sX = smem + (KD / 2) * WROW;    // [128][XROW]

    const int tid = threadIdx.x;
    const int lane = tid & 31;
    const int wave = tid >> 5;
    const int lrow = lane & 15;    // M (for A) / N (for B) within 16-wide tile
    const int lhalf = lane >> 4;   // selects K pair {k0,k0+1} vs {k0+2,k0+3}
    const int row0 = blockIdx.x * 128;

    // ---- stage W swizzled: pairs (W[k][n], W[k+1][n]) adjacent in LDS ----
    for (int i = tid * 4; i < KD * BN; i += 256 * 4) {
        float4 w = *(const float4*)(W + i);          // 4 consecutive n, same k
        int k = i / BN, n = i % BN;
        float* d = sW + (k >> 1) * WROW + 2 * n + (k & 1);
        d[0] = w.x; d[2] = w.y; d[4] = w.z; d[6] = w.w;
    }

    // ---- stage 128-row X strip: async global->LDS B128 (tracked by ASYNCcnt) ----
    const unsigned ldsX = (unsigned)(uintptr_t)sX;   // low 32 bits = LDS byte offset
    for (int i = tid * 4; i < 128 * KD; i += 256 * 4) {
        int r = i / KD, c = i % KD;
        int gr = row0 + r;
        unsigned loff = ldsX + (unsigned)(r * XROW + c) * 4u;
        if (gr < M) {
            unsigned goff = (unsigned)(((size_t)gr * KD + c) * 4u);
            asm volatile("global_load_async_to_lds_b128 %0, %1, %2"
                         :: "v"(loff), "v"(goff), "s"(A)
                         : "memory");
        } else {
            float* d = sX + r * XROW + c;
            d[0] = 0.f; d[1] = 0.f; d[2] = 0.f; d[3] = 0.f;
        }
    }
    asm volatile("s_wait_asynccnt 0x0" ::: "memory");
    __syncthreads();

    v8f acc[BN / 16];
    v8f zero = {0.f, 0.f, 0.f, 0.f, 0.f, 0.f, 0.f, 0.f};
#pragma unroll
    for (int t = 0; t < BN / 16; ++t) acc[t] = zero;

    const float* xrow = sX + (wave * 16 + lrow) * XROW;
#pragma unroll 4
    for (int k0 = 0; k0 < KD; k0 += 4) {
        const int kk = k0 + lhalf * 2;               // even
        const float* wbase = sW + (kk >> 1) * WROW + 2 * lrow;

        // batch-load A fragment + all B fragments for this k-step (all ds_load_b64)
        v2f a = *(const v2f*)(xrow + kk);            // a[j] = A[lane%16][kk + j]
        v2f bf[BN / 16];
#pragma unroll
        for (int t = 0; t < BN / 16; ++t)
            bf[t] = *(const v2f*)(wbase + 32 * t);   // b[j] = W[kk + j][t*16 + lane%16]

        // WMMA chain consumes fragments in REVERSE load order -> all bf[] live at
        // once -> loads batch ahead of a single dscnt wait before the chain.
#pragma unroll
        for (int t = BN / 16 - 1; t >= 0; --t)
            acc[t] = __builtin_amdgcn_wmma_f32_16x16x4_f32(
                false, a, false, bf[t], (short)0, acc[t], false, false);
    }

    // C/D layout: VGPR r -> row M=r (lanes 0-15), M=r+8 (lanes 16-31); col = lane%16
    const int grow0 = row0 + wave * 16 + lhalf * 8;
#pragma unroll
    for (int t = 0; t < BN / 16; ++t) {
#pragma unroll
        for (int r = 0; r < 8; ++r) {
            int gr = grow0 + r;
            if (gr < M) C[(size_t)gr * BN + t * 16 + lrow] = acc[t][r];
        }
    }
}

// ---------------- edge scatter: acc[dst] += dinv[src] * h[src] ----------------
template <int F>
__global__ void k_scatter(const int* __restrict__ ei, const float* __restrict__ h,
                          const float* __restrict__ dinv, float* __restrict__ acc) {
    int e = blockIdx.x * (blockDim.x >> 5) + (threadIdx.x >> 5);
    if (e >= NE) return;
    int lane = threadIdx.x & 31;
    int src = ei[e];
    int dst = ei[NE + e];
    float w = dinv[src];
    const float* hs = h + (size_t)src * F;
    float* ad = acc + (size_t)dst * F;
#pragma unroll
    for (int f = lane; f < F; f += 32)
        unsafeAtomicAdd(ad + f, w * hs[f]);
}

// ---------------- epilogue: a = relu(dinv*acc + dinv^2*h + b), in place over h ----
template <int F>
__global__ void k_epilogue(const float* __restrict__ acc, const float* __restrict__ dinv,
                           const float* __restrict__ b, float* __restrict__ h) {
    int i = blockIdx.x * blockDim.x + threadIdx.x;
    if (i >= NN * F) return;
    int node = i / F, f = i - node * F;
    float di = dinv[node];
    float v = di * acc[i] + di * di * h[i] + b[f];
    h[i] = fmaxf(v, 0.0f);
}

// ---------------- final: out = a2 @ W3 + b3 ----------------
__global__ void k_final(const float* __restrict__ a2, const float* __restrict__ W3,
                        const float* __restrict__ b3, float* __restrict__ out) {
    int i = blockIdx.x * blockDim.x + threadIdx.x;
    if (i >= NN) return;
    const float* r = a2 + (size_t)i * DH2;
    float s = 0.f;
#pragma unroll
    for (int j = 0; j < DH2; ++j) s += r[j] * W3[j];
    out[i] = s + b3[0];
}

extern "C" void kernel_launch(void* const* d_in, const int* in_sizes, int n_in,
                              void* d_out, int out_size, void* d_ws, size_t ws_size,
                              hipStream_t stream) {
    const float* x  = (const float*)d_in[0];
    const int*   ei = (const int*)d_in[1];
    const float* W1 = (const float*)d_in[2];
    const float* b1 = (const float*)d_in[3];
    const float* W2 = (const float*)d_in[4];
    const float* b2 = (const float*)d_in[5];
    const float* W3 = (const float*)d_in[6];
    const float* b3 = (const float*)d_in[7];
    float* out = (float*)d_out;

    // workspace carve-up (floats): dinv | bufB (N*128: h1 then a1) | bufC (N*128: acc1, then h2|acc2)
    float* dinv = (float*)d_ws;
    float* bufB = dinv + ((NN + 255) & ~255);
    float* bufC = bufB + (size_t)NN * DH1;
    float* h2   = bufC;
    float* acc2 = bufC + (size_t)NN * DH2;

    const int T = 256;

    // symmetric GCN normalization: deg (with self loop) -> dinv
    k_fill<<<cdiv(NN, T), T, 0, stream>>>(dinv, NN, 1.0f);
    k_deg<<<cdiv(NE, T), T, 0, stream>>>(ei, dinv);
    k_dinv<<<cdiv(NN, T), T, 0, stream>>>(dinv);

    // layer 1
    size_t lds1 = (size_t)((DIN / 2) * (2 * DH1 + 2) + 128 * (DIN + XPAD)) * sizeof(float);
    k_gemm_wmma<DH1, DIN><<<cdiv(NN, 128), T, lds1, stream>>>(x, W1, bufB, NN);
    k_fill<<<cdiv((long long)NN * DH1, T), T, 0, stream>>>(bufC, NN * DH1, 0.0f);
    k_scatter<DH1><<<cdiv(NE, 8), T, 0, stream>>>(ei, bufB, dinv, bufC);
    k_epilogue<DH1><<<cdiv((long long)NN * DH1, T), T, 0, stream>>>(bufC, dinv, b1, bufB);

    // layer 2
    size_t lds2 = (size_t)((DH1 / 2) * (2 * DH2 + 2) + 128 * (DH1 + XPAD)) * sizeof(float);
    k_gemm_wmma<DH2, DH1><<<cdiv(NN, 128), T, lds2, stream>>>(bufB, W2, h2, NN);
    k_fill<<<cdiv((long long)NN * DH2, T), T, 0, stream>>>(acc2, NN * DH2, 0.0f);
    k_scatter<DH2><<<cdiv(NE, 8), T, 0, stream>>>(ei, h2, dinv, acc2);
    k_epilogue<DH2><<<cdiv((long long)NN * DH2, T), T, 0, stream>>>(acc2, dinv, b2, h2);

    // output layer
    k_final<<<cdiv(NN, T), T, 0, stream>>>(h2, W3, b3, out);
}